// SwitchConv2d_5033701671496
// MI455X (gfx1250) — compile-verified
//
#include <hip/hip_runtime.h>

typedef __attribute__((ext_vector_type(2))) float v2f;
typedef __attribute__((ext_vector_type(8))) float v8f;

#define NB    256            // batch
#define NE    16             // experts
#define NO    128            // out channels
#define NI    128            // in channels
#define PPE   (NO * NI * 9)  // 147456 params per expert
#define NW_TOT (NE * 9 * NI * NO)  // 2359296 transformed weights
#define NBIAS  (NE * NO)           // 2048 combined biases

// ---------------------------------------------------------------------------
// Prep: combined weights in K-pair-interleaved layout wt[e][r][ip][o][2]
// (ip = i/2, last index = i%2), plus combined bias eb[e][o].
// A-fragment for (ci, ci+1) is then ONE aligned 64-bit load per lane.
// ---------------------------------------------------------------------------
__global__ __launch_bounds__(256) void prep_kernel(
    const float* __restrict__ weight, const float* __restrict__ weight_fact,
    const float* __restrict__ bias, const float* __restrict__ bias_fact,
    float* __restrict__ wt, float* __restrict__ eb) {
  int idx = blockIdx.x * blockDim.x + threadIdx.x;
  if (idx < NW_TOT) {
    // idx = (((e*9 + r)*64 + ip)*128 + o)*2 + j     (16384 = 64*128*2 = 2^14)
    int j  = idx & 1;
    int o  = (idx >> 1) & 127;
    int ip = (idx >> 8) & 63;
    int g  = idx >> 14;          // e*9 + r
    int e  = g / 9;
    int r  = g - e * 9;
    int i  = (ip << 1) + j;
    int p  = (o * NI + i) * 9 + r;
    wt[idx] = weight[e * PPE + p] + weight_fact[p];
  } else if (idx < NW_TOT + NBIAS) {
    int d = idx - NW_TOT;
    eb[d] = bias[d] + bias_fact[d & 127];
  }
}

// ---------------------------------------------------------------------------
// Main: one block per sample, 8 waves. Wave w owns M-tile w (16 out chans)
// x 4 N-tiles (64 spatial). K-loop: 9 kernel taps x 32 steps of K=4 on
// V_WMMA_F32_16X16X4_F32 (exact fp32 on the matrix pipe).
// LDS holds the padded sample in K-pair-interleaved form lx[ip][sp][2] so
// each B fragment is one aligned ds_load_b64.
// ---------------------------------------------------------------------------
__global__ __launch_bounds__(256) void switch_conv_kernel(
    const float* __restrict__ wt, const float* __restrict__ eb,
    const float* __restrict__ x, const int* __restrict__ route,
    float* __restrict__ out) {
  __shared__ float lx[NI * 100];  // 64 pairs * 100 padded positions * 2

  const int b   = blockIdx.x;
  const int tid = threadIdx.x;
  const int e   = route[b];

  // Stage padded input (10x10 halo-zeroed) into pair-interleaved LDS.
  const float* xb = x + (size_t)b * NI * 64;
  for (int idx = tid; idx < NI * 100; idx += 256) {
    int i  = idx / 100;
    int sp = idx - i * 100;
    int y  = sp / 10;
    int xx = sp - y * 10;
    float v = 0.0f;
    if (y >= 1 && y <= 8 && xx >= 1 && xx <= 8)
      v = xb[i * 64 + (y - 1) * 8 + (xx - 1)];
    lx[(i >> 1) * 200 + sp * 2 + (i & 1)] = v;
  }
  __syncthreads();

  const int lane  = tid & 31;
  const int wave  = tid >> 5;
  const int nlo   = lane & 15;     // N (and M) column-within-tile
  const int khalf = lane >> 4;     // selects K sub-pair (0 or 1)
  const int mbase = wave << 4;     // 16 output channels per wave

  const v2f* lxp = (const v2f*)lx;       // lxp[ip*100 + sp]
  const v2f* wp0 = (const v2f*)wt;       // wp0[((e*9+r)*64 + ip)*128 + o]

  v8f acc0 = {}, acc1 = {}, acc2 = {}, acc3 = {};

  for (int r = 0; r < 9; ++r) {
    const int kh = r / 3;
    const int kw = r - kh * 3;

    // per-lane spatial offsets (padded coords) for the 4 N-tiles
    int s0[4];
#pragma unroll
    for (int nt = 0; nt < 4; ++nt) {
      int p  = (nt << 4) + nlo;    // spatial position 0..63
      int y  = (p >> 3) + kh;      // padded row 0..9
      int xx = (p & 7) + kw;       // padded col 0..9
      s0[nt] = y * 10 + xx;
    }

    const v2f* wr = wp0 + (((size_t)(e * 9 + r)) << 13) + mbase + nlo;

#pragma unroll 4
    for (int ib = 0; ib < NI; ib += 4) {
      const int ip = (ib >> 1) + khalf;     // K-pair index
      v2f a  = wr[(size_t)ip << 7];         // one global_load_b64, coalesced
      const int lb = ip * 100;
      v2f b0 = lxp[lb + s0[0]];             // one ds_load_b64 each
      v2f b1 = lxp[lb + s0[1]];
      v2f b2 = lxp[lb + s0[2]];
      v2f b3 = lxp[lb + s0[3]];

      acc0 = __builtin_amdgcn_wmma_f32_16x16x4_f32(false, a, false, b0,
                                                   (short)0, acc0, false, false);
      acc1 = __builtin_amdgcn_wmma_f32_16x16x4_f32(false, a, false, b1,
                                                   (short)0, acc1, false, false);
      acc2 = __builtin_amdgcn_wmma_f32_16x16x4_f32(false, a, false, b2,
                                                   (short)0, acc2, false, false);
      acc3 = __builtin_amdgcn_wmma_f32_16x16x4_f32(false, a, false, b3,
                                                   (short)0, acc3, false, false);
    }
  }

  // Epilogue: C/D layout (VGPR rr -> M = rr + 8*khalf, lane -> N) + bias.
  const int khalf8 = khalf << 3;
  float* ob = out + ((size_t)b * NO + mbase) * 64;
#pragma unroll
  for (int rr = 0; rr < 8; ++rr) {
    int orow = rr + khalf8;
    float bi = eb[e * NO + mbase + orow];
    ob[orow * 64 +  0 + nlo] = acc0[rr] + bi;
    ob[orow * 64 + 16 + nlo] = acc1[rr] + bi;
    ob[orow * 64 + 32 + nlo] = acc2[rr] + bi;
    ob[orow * 64 + 48 + nlo] = acc3[rr] + bi;
  }
}

// ---------------------------------------------------------------------------
extern "C" void kernel_launch(void* const* d_in, const int* in_sizes, int n_in,
                              void* d_out, int out_size, void* d_ws, size_t ws_size,
                              hipStream_t stream) {
  const float* x           = (const float*)d_in[0];
  const int*   route       = (const int*)d_in[1];
  const float* weight      = (const float*)d_in[2];
  const float* weight_fact = (const float*)d_in[3];
  const float* bias        = (const float*)d_in[4];
  const float* bias_fact   = (const float*)d_in[5];

  float* wt = (float*)d_ws;       // 2359296 floats: pair-interleaved weights
  float* eb = wt + NW_TOT;        // 2048 floats: combined bias

  int total = NW_TOT + NBIAS;
  prep_kernel<<<(total + 255) / 256, 256, 0, stream>>>(weight, weight_fact,
                                                       bias, bias_fact, wt, eb);
  switch_conv_kernel<<<NB, 256, 0, stream>>>(wt, eb, x, route, (float*)d_out);
}